// GCNEncoder_59708635349040
// MI455X (gfx1250) — compile-verified
//
#include <hip/hip_runtime.h>
#include <hip/hip_bf16.h>

typedef float v2f __attribute__((ext_vector_type(2)));
typedef float v8f __attribute__((ext_vector_type(8)));

#define FD 128  // feature dim

// ---------------------------------------------------------------------------
// Degree / normalization
// ---------------------------------------------------------------------------
__global__ __launch_bounds__(256) void gcn_deg(const int* __restrict__ dst,
                                               float* __restrict__ deg, int E_) {
  int i = blockIdx.x * 256 + threadIdx.x;
  if (i < E_) atomicAdd(&deg[dst[i]], 1.0f);
}

__global__ __launch_bounds__(256) void gcn_dinv(float* __restrict__ deg, int n_) {
  int i = blockIdx.x * 256 + threadIdx.x;
  if (i < n_) deg[i] = rsqrtf(deg[i] + 1.0f);  // +1 self-loop; always >= 1
}

// ---------------------------------------------------------------------------
// GEMM: T[N x 128] = H[N x 128] * W[128 x 128] via V_WMMA_F32_16X16X4_F32.
// One wave -> 16 rows x 128 cols (8 v8f accumulators). W staged through LDS
// in 16-row K-tiles shared by the 8 waves of the block, stored K-pair
// interleaved: Wlds[(k/2)*256 + c*2 + (k&1)] so a lane's {W[k][c], W[k+1][c]}
// B-fragment is one aligned ds_load_b64 (no register re-pairing moves).
// ---------------------------------------------------------------------------
__global__ __launch_bounds__(256) void gcn_gemm(const float* __restrict__ H,
                                                const float* __restrict__ W,
                                                float* __restrict__ T,
                                                int nStrips) {
  __shared__ float Wlds[16 * FD];  // 8 KB K-tile of W (pair-interleaved)
  const int tid  = threadIdx.x;
  const int wave = tid >> 5;
  const int lane = tid & 31;

  int strip = blockIdx.x * 8 + wave;
  const bool valid = (strip < nStrips);
  if (!valid) strip = nStrips - 1;  // clamp: keep loads in-bounds, skip stores

  const int row0  = strip * 16;
  const int r     = lane & 15;          // A: row in tile / B: col in tile
  const int khalf = (lane >> 4) * 2;    // 0 or 2 (lane-half K split)

  v8f acc[8];
#pragma unroll
  for (int n = 0; n < 8; ++n) acc[n] = (v8f)(0.0f);

  for (int kt = 0; kt < 8; ++kt) {      // 8 K-tiles of 16
    __syncthreads();
#pragma unroll
    for (int i = 0; i < 8; ++i) {       // 256 thr * 8 = 2048 floats
      int off = i * 256 + tid;          // element (k,c) of this K-tile
      int k = off >> 7;
      int c = off & (FD - 1);
      Wlds[(k >> 1) * 256 + c * 2 + (k & 1)] = W[kt * 16 * FD + off];
    }
    __syncthreads();

#pragma unroll
    for (int kk = 0; kk < 16; kk += 4) {
      const int k0 = kt * 16 + kk;
      // A fragment: 16x4 fp32 (ISA 7.12.2): lane r holds K = khalf, khalf+1
      v2f a = *(const v2f*)(H + (size_t)(row0 + r) * FD + (k0 + khalf));
      const float* bbase = &Wlds[((kk + khalf) >> 1) * 256 + r * 2];
#pragma unroll
      for (int n = 0; n < 8; ++n) {
        // B fragment: 4x16 fp32, lane r = column n*16+r, lane-half splits K
        v2f b = *(const v2f*)(bbase + n * 32);
        acc[n] = __builtin_amdgcn_wmma_f32_16x16x4_f32(
            false, a, false, b, (short)0, acc[n], false, false);
      }
    }
  }

  if (valid) {
    const int rowAdd = (lane >> 4) * 8;  // C/D: lanes 16-31 hold M = v+8
#pragma unroll
    for (int n = 0; n < 8; ++n) {
#pragma unroll
      for (int v = 0; v < 8; ++v) {
        T[(size_t)(row0 + v + rowAdd) * FD + n * 16 + r] = acc[n][v];
      }
    }
  }
}

// ---------------------------------------------------------------------------
// agg = bias + self-loop contribution (t[i] * dinv[i]^2)
// ---------------------------------------------------------------------------
__global__ __launch_bounds__(256) void gcn_init_agg(const float* __restrict__ T,
                                                    const float* __restrict__ dinv,
                                                    const float* __restrict__ bias,
                                                    float* __restrict__ agg,
                                                    size_t total) {
  size_t i = (size_t)blockIdx.x * 256 + threadIdx.x;
  if (i >= total) return;
  int node = (int)(i >> 7);
  int f = (int)(i & (FD - 1));
  float di = dinv[node];
  agg[i] = bias[f] + T[i] * di * di;
}

// ---------------------------------------------------------------------------
// Edge scatter: one wave per edge, lane handles 4 features (float4 gather
// from L2-resident T, 4 float atomics into agg).
// ---------------------------------------------------------------------------
__global__ __launch_bounds__(256) void gcn_scatter(const float* __restrict__ T,
                                                   const int* __restrict__ src,
                                                   const int* __restrict__ dst,
                                                   const float* __restrict__ dinv,
                                                   float* __restrict__ agg, int E_) {
  int e = blockIdx.x * 8 + (threadIdx.x >> 5);
  if (e >= E_) return;
  int lane = threadIdx.x & 31;
  int s = src[e], d = dst[e];
  float nrm = dinv[s] * dinv[d];
  float4 v = ((const float4*)(T + (size_t)s * FD))[lane];
  float* out = agg + (size_t)d * FD + lane * 4;
  atomicAdd(out + 0, v.x * nrm);
  atomicAdd(out + 1, v.y * nrm);
  atomicAdd(out + 2, v.z * nrm);
  atomicAdd(out + 3, v.w * nrm);
}

// ---------------------------------------------------------------------------
// BatchNorm: per-column sum / sumsq partials, then fused normalize + ReLU.
// ---------------------------------------------------------------------------
__global__ __launch_bounds__(128) void gcn_bn_stats(const float* __restrict__ agg,
                                                    float* __restrict__ sums,
                                                    int n_, int rowsPerBlock) {
  int col = threadIdx.x;
  int r0 = blockIdx.x * rowsPerBlock;
  int r1 = r0 + rowsPerBlock;
  if (r1 > n_) r1 = n_;
  float s = 0.0f, s2 = 0.0f;
  for (int row = r0; row < r1; ++row) {
    float v = agg[(size_t)row * FD + col];
    s += v;
    s2 += v * v;
  }
  atomicAdd(&sums[col], s);
  atomicAdd(&sums[FD + col], s2);
}

__global__ __launch_bounds__(256) void gcn_bn_apply(const float* __restrict__ sums,
                                                    const float* __restrict__ gamma,
                                                    const float* __restrict__ beta,
                                                    float* __restrict__ h,
                                                    size_t total, float invN) {
  size_t i = (size_t)blockIdx.x * 256 + threadIdx.x;
  if (i >= total) return;
  int f = (int)(i & (FD - 1));
  float mu = sums[f] * invN;
  float var = sums[FD + f] * invN - mu * mu;
  float v = gamma[f] * (h[i] - mu) * rsqrtf(var + 1e-5f) + beta[f];
  h[i] = fmaxf(v, 0.0f);
}

// ---------------------------------------------------------------------------
// global_add_pool: one wave per node, lane handles 4 features.
// ---------------------------------------------------------------------------
__global__ __launch_bounds__(256) void gcn_pool(const float* __restrict__ h,
                                                const int* __restrict__ batch,
                                                float* __restrict__ pool, int n_) {
  int node = blockIdx.x * 8 + (threadIdx.x >> 5);
  if (node >= n_) return;
  int lane = threadIdx.x & 31;
  int g = batch[node];
  float4 v = ((const float4*)(h + (size_t)node * FD))[lane];
  float* out = pool + (size_t)g * FD + lane * 4;
  atomicAdd(out + 0, v.x);
  atomicAdd(out + 1, v.y);
  atomicAdd(out + 2, v.z);
  atomicAdd(out + 3, v.w);
}

// ---------------------------------------------------------------------------
extern "C" void kernel_launch(void* const* d_in, const int* in_sizes, int n_in,
                              void* d_out, int out_size, void* d_ws, size_t ws_size,
                              hipStream_t stream) {
  const float* x      = (const float*)d_in[0];
  const int*   edge   = (const int*)d_in[1];
  const int*   batch  = (const int*)d_in[2];
  const float* Ws     = (const float*)d_in[3];
  const float* bs     = (const float*)d_in[4];
  const float* gammas = (const float*)d_in[5];
  const float* betas  = (const float*)d_in[6];

  const int N_ = in_sizes[0] / FD;            // 100000
  const int E_ = in_sizes[1] / 2;             // 1600000
  const int L_ = in_sizes[3] / (FD * FD);     // 3
  const int G_ = out_size / FD - N_;          // 128

  const int* src = edge;        // edge_index[0]
  const int* dst = edge + E_;   // edge_index[1]

  float* out  = (float*)d_out;
  float* h    = out;                          // [N, 128] (also agg buffer)
  float* pool = out + (size_t)N_ * FD;        // [G, 128]

  float* T    = (float*)d_ws;                 // [N, 128] scratch (t = h @ W)
  float* dinv = T + (size_t)N_ * FD;          // [N]
  float* sums = dinv + N_;                    // [256] sum | sumsq

  const size_t nd = (size_t)N_ * FD;

  // Degrees -> dinv
  hipMemsetAsync(dinv, 0, (size_t)N_ * sizeof(float), stream);
  gcn_deg<<<(E_ + 255) / 256, 256, 0, stream>>>(dst, dinv, E_);
  gcn_dinv<<<(N_ + 255) / 256, 256, 0, stream>>>(dinv, N_);

  const int nStrips = (N_ + 15) / 16;         // 6250 (N multiple of 16)

  for (int l = 0; l < L_; ++l) {
    const float* Hin = (l == 0) ? x : h;
    gcn_gemm<<<(nStrips + 7) / 8, 256, 0, stream>>>(
        Hin, Ws + (size_t)l * FD * FD, T, nStrips);
    // h buffer is free now (GEMM consumed it); reuse as agg.
    gcn_init_agg<<<(int)((nd + 255) / 256), 256, 0, stream>>>(
        T, dinv, bs + (size_t)l * FD, h, nd);
    gcn_scatter<<<(E_ + 7) / 8, 256, 0, stream>>>(T, src, dst, dinv, h, E_);
    hipMemsetAsync(sums, 0, 2 * FD * sizeof(float), stream);
    gcn_bn_stats<<<200, 128, 0, stream>>>(h, sums, N_, (N_ + 199) / 200);
    gcn_bn_apply<<<(int)((nd + 255) / 256), 256, 0, stream>>>(
        sums, gammas + (size_t)l * FD, betas + (size_t)l * FD, h, nd,
        1.0f / (float)N_);
  }

  hipMemsetAsync(pool, 0, (size_t)G_ * FD * sizeof(float), stream);
  gcn_pool<<<(N_ + 7) / 8, 256, 0, stream>>>(h, batch, pool, N_);
}